// Encoder_80693845557765
// MI455X (gfx1250) — compile-verified
//
#include <hip/hip_runtime.h>
#include <math.h>

#define DD 128          // encoder dim
#define PST 132         // LDS panel row stride (floats): kills stride-128 bank conflicts
#define WPB 2           // waves (16-node tiles) per block in mp_kernel
#define SSTR 20         // B-slab row stride (floats): 16 cols + 4 pad

typedef __attribute__((ext_vector_type(2))) float v2f;
typedef __attribute__((ext_vector_type(8))) float v8f;

__device__ __forceinline__ float sigm(float x) { return 1.0f / (1.0f + __expf(-x)); }
__device__ __forceinline__ float lrelu(float v) { return v > 0.0f ? v : 0.01f * v; }

__device__ __forceinline__ void atomicMaxF(float* addr, float val) {
    unsigned int* a = (unsigned int*)addr;
    unsigned int old = __float_as_uint(*addr);
    while (__uint_as_float(old) < val) {
        unsigned int assumed = old;
        old = atomicCAS(a, assumed, __float_as_uint(val));
        if (old == assumed) break;
    }
}

// ---------------------------------------------------------------- utilities
__global__ void fill_kernel(float* __restrict__ p, long long n, float v) {
    long long i = (long long)blockIdx.x * blockDim.x + threadIdx.x;
    if (i < n) p[i] = v;
}

// ------------------------------------------------- node embedding: 16->64->128
__global__ __launch_bounds__(128) void embed_kernel(
    const float* __restrict__ x, const float* __restrict__ W_emb,
    const float* __restrict__ b_emb, const float* __restrict__ W_first,
    const float* __restrict__ b_first, float* __restrict__ out, int nN)
{
    const int node = blockIdx.x;
    if (node >= nN) return;
    __shared__ float xv[16];
    __shared__ float evs[64];
    const int tid = threadIdx.x;
    if (tid < 16) xv[tid] = x[(size_t)node * 16 + tid];
    __syncthreads();
    if (tid < 64) {
        float s = b_emb[tid];
        #pragma unroll
        for (int k = 0; k < 16; ++k) s += xv[k] * W_emb[k * 64 + tid];
        evs[tid] = lrelu(s);
    }
    __syncthreads();
    float s = b_first[tid];
    for (int k = 0; k < 64; ++k) s += evs[k] * W_first[k * DD + tid];
    out[(size_t)node * DD + tid] = lrelu(s);
}

// ------------------------------------------------- edge scatter: agg[dst]+=out[src]
__global__ __launch_bounds__(256) void scatter_kernel(
    const float* __restrict__ out, const int* __restrict__ ei,
    float* __restrict__ agg, int nE)
{
    const int e = blockIdx.x * (blockDim.x >> 5) + (threadIdx.x >> 5);
    if (e >= nE) return;
    const int lane = threadIdx.x & 31;
    const int src = ei[e];
    const int dst = ei[nE + e];
    const float4 v = *(const float4*)(out + (size_t)src * DD + lane * 4);
    float* ap = agg + (size_t)dst * DD + lane * 4;
    atomicAdd(ap + 0, v.x);
    atomicAdd(ap + 1, v.y);
    atomicAdd(ap + 2, v.z);
    atomicAdd(ap + 3, v.w);
}

// ------------------------------------------------- fused SAGE + GRU step (WMMA f32)
// Each wave32 owns one 16-node tile; both waves of a block run the jn loop in
// lockstep and share one LDS-staged B column slab (128K x 16N).
// h == out on entry (loop invariant of the reference).
__global__ __launch_bounds__(32 * WPB) void mp_kernel(
    const float* __restrict__ agg, float* __restrict__ out,
    const float* __restrict__ W_l, const float* __restrict__ b_l,
    const float* __restrict__ W_r,
    const float* __restrict__ Wih, const float* __restrict__ Whh,
    const float* __restrict__ bih, const float* __restrict__ bhh,
    int n_tiles)
{
    __shared__ float p_out[WPB][16 * PST];
    __shared__ float p_agg[WPB][16 * PST];
    __shared__ float p_m  [WPB][16 * PST];
    __shared__ float bs[DD * SSTR];            // shared B slab: 128 x 20 floats

    const int tid  = threadIdx.x;
    const int wave = tid >> 5;
    const int lane = tid & 31;
    const int tile = blockIdx.x * WPB + wave;
    const bool valid = (tile < n_tiles);
    const int node0 = tile * 16;
    float* Pout = p_out[wave];
    float* Pagg = p_agg[wave];
    float* Pm   = p_m[wave];

    // Stage A panels into LDS (16 rows x 128 floats, padded stride)
    if (valid) {
        for (int idx = lane; idx < 16 * 32; idx += 32) {
            const int row = idx >> 5;
            const int c4  = (idx & 31) << 2;
            const float4 vo = *(const float4*)(out + (size_t)(node0 + row) * DD + c4);
            const float4 va = *(const float4*)(agg + (size_t)(node0 + row) * DD + c4);
            *(float4*)(Pout + row * PST + c4) = vo;
            *(float4*)(Pagg + row * PST + c4) = va;
        }
    }

    const int l15 = lane & 15;          // M (A) / N (B,C) index
    const int kh2 = (lane >> 4) << 1;   // K sub-offset: 0 or 2

    // Cooperative B slab staging: W[k][col0..col0+15], k = 0..127 -> bs[k*SSTR + c]
    auto stage_slab = [&](const float* __restrict__ W, int ldw, int col0) {
        for (int idx = tid; idx < 512; idx += 32 * WPB) {
            const int k = idx >> 2;
            const int c = (idx & 3) << 2;
            const float4 v = *(const float4*)(W + (size_t)k * ldw + col0 + c);
            *(float4*)(bs + k * SSTR + c) = v;
        }
    };

    // K=128 accumulation chain: A from a LDS panel, B from the shared slab.
    auto chain = [&](const float* __restrict__ Pa, v8f acc) -> v8f {
        const float* pa = Pa + l15 * PST + kh2;
        const float* pb = bs + kh2 * SSTR + l15;
        #pragma unroll
        for (int k0 = 0; k0 < DD; k0 += 4) {
            v2f a, b;
            a.x = pa[k0];             a.y = pa[k0 + 1];
            b.x = pb[k0 * SSTR];      b.y = pb[k0 * SSTR + SSTR];
            acc = __builtin_amdgcn_wmma_f32_16x16x4_f32(false, a, false, b,
                                                        (short)0, acc, false, false);
        }
        return acc;
    };

    // ---- stage 1: m = relu(agg @ W_l + b_l + out @ W_r) -> Pm
    for (int jn = 0; jn < 8; ++jn) {
        const int col = jn * 16 + l15;
        const float bias = b_l[col];
        v8f acc;
        #pragma unroll
        for (int i = 0; i < 8; ++i) acc[i] = bias;

        __syncthreads();
        stage_slab(W_l, DD, jn * 16);
        __syncthreads();
        if (valid) acc = chain(Pagg, acc);

        __syncthreads();
        stage_slab(W_r, DD, jn * 16);
        __syncthreads();
        if (valid) {
            acc = chain(Pout, acc);
            #pragma unroll
            for (int i = 0; i < 8; ++i) {
                const float v = acc[i] > 0.0f ? acc[i] : 0.0f;
                Pm[(i + 4 * kh2) * PST + col] = v;   // row = i + 8*(lane>>4)
            }
        }
    }

    // ---- stage 2: GRU. 6 accumulator tiles per jn (r,z,n for input & hidden).
    for (int jn = 0; jn < 8; ++jn) {
        const int colb = jn * 16;
        const int col  = colb + l15;
        v8f g[6];
        {
            const float b0 = bih[col], b1 = bih[DD + col], b2 = bih[2 * DD + col];
            const float c0 = bhh[col], c1 = bhh[DD + col], c2 = bhh[2 * DD + col];
            #pragma unroll
            for (int i = 0; i < 8; ++i) {
                g[0][i] = b0; g[1][i] = b1; g[2][i] = b2;
                g[3][i] = c0; g[4][i] = c1; g[5][i] = c2;
            }
        }
        #pragma unroll
        for (int s = 0; s < 6; ++s) {
            const float* W = (s < 3) ? Wih : Whh;
            const int wcol = (s % 3) * DD + colb;
            __syncthreads();
            stage_slab(W, 3 * DD, wcol);
            __syncthreads();
            if (valid) g[s] = chain((s < 3) ? Pm : Pout, g[s]);
        }
        if (valid) {
            #pragma unroll
            for (int i = 0; i < 8; ++i) {
                const float r  = sigm(g[0][i] + g[3][i]);
                const float z  = sigm(g[1][i] + g[4][i]);
                const float nn = tanhf(g[2][i] + r * g[5][i]);
                const float hv = Pout[(i + 4 * kh2) * PST + col];
                out[(size_t)(node0 + i + 4 * kh2) * DD + col] = (1.0f - z) * nn + z * hv;
            }
        }
    }
}

// ------------------------------------------------- Set2Set LSTM step (512 graphs)
__global__ __launch_bounds__(128) void lstm_kernel(
    const float* __restrict__ qs, const float* __restrict__ hin,
    const float* __restrict__ cin, float* __restrict__ hout, float* __restrict__ cout_,
    const float* __restrict__ Wih, const float* __restrict__ Whh,
    const float* __restrict__ bih, const float* __restrict__ bhh)
{
    const int b = blockIdx.x;
    const int j = threadIdx.x;
    __shared__ float sq[256];
    __shared__ float sh[128];
    sq[j] = qs[b * 256 + j];
    sq[128 + j] = qs[b * 256 + 128 + j];
    sh[j] = hin[b * 128 + j];
    __syncthreads();
    float g[4];
    #pragma unroll
    for (int gate = 0; gate < 4; ++gate) {
        const int c = gate * 128 + j;
        float s = bih[c] + bhh[c];
        for (int k = 0; k < 256; ++k) s += sq[k] * Wih[k * 512 + c];
        for (int k = 0; k < 128; ++k) s += sh[k] * Whh[k * 512 + c];
        g[gate] = s;
    }
    const float gi = sigm(g[0]), gf = sigm(g[1]), gg = tanhf(g[2]), go = sigm(g[3]);
    const float c_new = gf * cin[b * 128 + j] + gi * gg;
    hout[b * 128 + j] = go * tanhf(c_new);
    cout_[b * 128 + j] = c_new;
}

// ------------------------------------------------- attention e = <out, hh[batch]>, segment max
__global__ __launch_bounds__(256) void attn_e_kernel(
    const float* __restrict__ out, const float* __restrict__ hh,
    const int* __restrict__ batch, float* __restrict__ ev, float* __restrict__ emax, int nN)
{
    const int node = blockIdx.x * (blockDim.x >> 5) + (threadIdx.x >> 5);
    if (node >= nN) return;
    const int lane = threadIdx.x & 31;
    const int b = batch[node];
    const float4 vo = *(const float4*)(out + (size_t)node * DD + lane * 4);
    const float4 vh = *(const float4*)(hh + (size_t)b * DD + lane * 4);
    float p = vo.x * vh.x + vo.y * vh.y + vo.z * vh.z + vo.w * vh.w;
    #pragma unroll
    for (int off = 16; off > 0; off >>= 1) p += __shfl_xor(p, off, 32);
    if (lane == 0) {
        ev[node] = p;
        atomicMaxF(&emax[b], p);
    }
}

__global__ void attn_a_kernel(const float* __restrict__ ev, const float* __restrict__ emax,
                              const int* __restrict__ batch, float* __restrict__ av,
                              float* __restrict__ ssum, int nN)
{
    const int i = blockIdx.x * blockDim.x + threadIdx.x;
    if (i >= nN) return;
    const int b = batch[i];
    const float a = expf(ev[i] - emax[b]);
    av[i] = a;
    atomicAdd(&ssum[b], a);
}

__global__ __launch_bounds__(256) void attn_r_kernel(
    const float* __restrict__ av, const float* __restrict__ ssum,
    const int* __restrict__ batch, const float* __restrict__ out,
    float* __restrict__ rv, int nN)
{
    const int node = blockIdx.x * (blockDim.x >> 5) + (threadIdx.x >> 5);
    if (node >= nN) return;
    const int lane = threadIdx.x & 31;
    const int b = batch[node];
    const float w = av[node] / (ssum[b] + 1e-16f);
    const float4 vo = *(const float4*)(out + (size_t)node * DD + lane * 4);
    float* rp = rv + (size_t)b * DD + lane * 4;
    atomicAdd(rp + 0, w * vo.x);
    atomicAdd(rp + 1, w * vo.y);
    atomicAdd(rp + 2, w * vo.z);
    atomicAdd(rp + 3, w * vo.w);
}

__global__ void qs_kernel(const float* __restrict__ hh, const float* __restrict__ rv,
                          float* __restrict__ qs)
{
    const int b = blockIdx.x, j = threadIdx.x;
    qs[b * 256 + j] = hh[b * 128 + j];
    qs[b * 256 + 128 + j] = rv[b * 128 + j];
}

__global__ __launch_bounds__(128) void lin1_kernel(
    const float* __restrict__ qs, const float* __restrict__ W,
    const float* __restrict__ bias, float* __restrict__ t1)
{
    const int b = blockIdx.x, j = threadIdx.x;
    __shared__ float sq[256];
    sq[j] = qs[b * 256 + j];
    sq[128 + j] = qs[b * 256 + 128 + j];
    __syncthreads();
    float s = bias[j];
    for (int k = 0; k < 256; ++k) s += sq[k] * W[k * 128 + j];
    t1[b * 128 + j] = lrelu(s);
}

__global__ __launch_bounds__(64) void lin2_kernel(
    const float* __restrict__ t1, const float* __restrict__ W,
    const float* __restrict__ bias, float* __restrict__ outF)
{
    const int b = blockIdx.x, j = threadIdx.x;
    __shared__ float st[128];
    st[j] = t1[b * 128 + j];
    st[64 + j] = t1[b * 128 + 64 + j];
    __syncthreads();
    float s = bias[j];
    for (int k = 0; k < 128; ++k) s += st[k] * W[k * 64 + j];
    outF[b * 64 + j] = s;
}

extern "C" void kernel_launch(void* const* d_in, const int* in_sizes, int n_in,
                              void* d_out, int out_size, void* d_ws, size_t ws_size,
                              hipStream_t stream)
{
    (void)n_in; (void)ws_size;
    const float* x       = (const float*)d_in[0];
    const int*   ei      = (const int*)d_in[1];
    /* d_in[2] edge_attr unused by reference */
    const int*   batch   = (const int*)d_in[3];
    const float* W_emb   = (const float*)d_in[4];
    const float* b_emb   = (const float*)d_in[5];
    const float* W_first = (const float*)d_in[6];
    const float* b_first = (const float*)d_in[7];
    const float* W_l     = (const float*)d_in[8];
    const float* b_l     = (const float*)d_in[9];
    const float* W_r     = (const float*)d_in[10];
    const float* gWih    = (const float*)d_in[11];
    const float* gWhh    = (const float*)d_in[12];
    const float* gbih    = (const float*)d_in[13];
    const float* gbhh    = (const float*)d_in[14];
    const float* lWih    = (const float*)d_in[15];
    const float* lWhh    = (const float*)d_in[16];
    const float* lbih    = (const float*)d_in[17];
    const float* lbhh    = (const float*)d_in[18];
    const float* W_lin   = (const float*)d_in[19];
    const float* b_lin   = (const float*)d_in[20];
    const float* W_fin   = (const float*)d_in[21];
    const float* b_fin   = (const float*)d_in[22];

    const int N = in_sizes[0] / 16;   // 50000 nodes
    const int E = in_sizes[1] / 2;    // 800000 edges
    const int B = out_size / 64;      // 512 graphs

    float* ws   = (float*)d_ws;
    float* out  = ws; ws += (size_t)N * DD;
    float* agg  = ws; ws += (size_t)N * DD;
    float* ev   = ws; ws += N;
    float* av   = ws; ws += N;
    float* emax = ws; ws += B;
    float* ssum = ws; ws += B;
    float* rv   = ws; ws += (size_t)B * DD;
    float* qs   = ws; ws += (size_t)B * 2 * DD;
    float* hhA  = ws; ws += (size_t)B * DD;
    float* hhB  = ws; ws += (size_t)B * DD;
    float* ccA  = ws; ws += (size_t)B * DD;
    float* ccB  = ws; ws += (size_t)B * DD;
    float* t1   = ws; ws += (size_t)B * DD;

    // 1) node embedding MLP
    embed_kernel<<<N, 128, 0, stream>>>(x, W_emb, b_emb, W_first, b_first, out, N);

    // 2) three message-passing steps (h == out invariant lets us keep one state array)
    const int n_tiles = N / 16;
    const long long nAgg = (long long)N * DD;
    for (int step = 0; step < 3; ++step) {
        fill_kernel<<<(int)((nAgg + 255) / 256), 256, 0, stream>>>(agg, nAgg, 0.0f);
        scatter_kernel<<<(E + 7) / 8, 256, 0, stream>>>(out, ei, agg, E);
        mp_kernel<<<(n_tiles + WPB - 1) / WPB, 32 * WPB, 0, stream>>>(
            agg, out, W_l, b_l, W_r, gWih, gWhh, gbih, gbhh, n_tiles);
    }

    // 3) Set2Set
    fill_kernel<<<(B * 2 * DD + 255) / 256, 256, 0, stream>>>(qs, B * 2 * DD, 0.0f);
    fill_kernel<<<(B * DD + 255) / 256, 256, 0, stream>>>(hhA, B * DD, 0.0f);
    fill_kernel<<<(B * DD + 255) / 256, 256, 0, stream>>>(ccA, B * DD, 0.0f);
    for (int t = 0; t < 3; ++t) {
        float* hin   = (t & 1) ? hhB : hhA;
        float* hout  = (t & 1) ? hhA : hhB;
        float* cin   = (t & 1) ? ccB : ccA;
        float* cout_ = (t & 1) ? ccA : ccB;
        lstm_kernel<<<B, 128, 0, stream>>>(qs, hin, cin, hout, cout_, lWih, lWhh, lbih, lbhh);
        fill_kernel<<<(B + 255) / 256, 256, 0, stream>>>(emax, B, -3.0e38f);
        fill_kernel<<<(B + 255) / 256, 256, 0, stream>>>(ssum, B, 0.0f);
        fill_kernel<<<(B * DD + 255) / 256, 256, 0, stream>>>(rv, B * DD, 0.0f);
        attn_e_kernel<<<(N + 7) / 8, 256, 0, stream>>>(out, hout, batch, ev, emax, N);
        attn_a_kernel<<<(N + 255) / 256, 256, 0, stream>>>(ev, emax, batch, av, ssum, N);
        attn_r_kernel<<<(N + 7) / 8, 256, 0, stream>>>(av, ssum, batch, out, rv, N);
        qs_kernel<<<B, 128, 0, stream>>>(hout, rv, qs);
    }

    // 4) head
    lin1_kernel<<<B, 128, 0, stream>>>(qs, W_lin, b_lin, t1);
    lin2_kernel<<<B, 64, 0, stream>>>(t1, W_fin, b_fin, (float*)d_out);
}